// Group_21904333209874
// MI455X (gfx1250) — compile-verified
//
#include <hip/hip_runtime.h>
#include <hip/hip_bf16.h>

typedef float v2f __attribute__((ext_vector_type(2)));
typedef float v8f __attribute__((ext_vector_type(8)));

#define FPS_THREADS 1024
#define KNN_THREADS 512
#define CHUNK 512
#define SROW 17   // 512x16 score tile padded to 17 floats/row -> conflict-free column reads

// uniform-lane broadcasts: v_readlane_b32 (no LDS round-trip, result in SGPR)
static __device__ __forceinline__ float readlane_f32(float v, int lane) {
    return __int_as_float(__builtin_amdgcn_readlane(__float_as_int(v), lane));
}

// ---------------- prep: pack float4 {x,y,z,||x||^2} ----------------
__global__ void prep_kernel(const float* __restrict__ xyz, float4* __restrict__ pts4, int N) {
    int i = blockIdx.x * blockDim.x + threadIdx.x;
    if (i < N) {
        float x = xyz[3*i+0], y = xyz[3*i+1], z = xyz[3*i+2];
        pts4[i] = make_float4(x, y, z, x*x + y*y + z*z);
    }
}

// ---------------- FPS: one workgroup, dists resident in LDS ----------------
__global__ __launch_bounds__(FPS_THREADS) void fps_kernel(const float4* __restrict__ pts4,
                                                          int* __restrict__ fpsIdx,
                                                          int N, int G) {
    extern __shared__ float smem[];
    float* sd    = smem;                 // N dists
    float* wv    = smem + N;             // 32 wave partial max vals
    int*   wi    = (int*)(wv + 32);      // 32 wave partial max idxs
    int*   s_nxt = wi + 32;              // broadcast slot

    const int t    = threadIdx.x;
    const int lane = t & 31;
    const int wid  = t >> 5;
    const int PT   = N / FPS_THREADS;    // 32 points per thread

    float4 p0 = pts4[0];
    float mv = -1.0f; int mi = 0;
    for (int k = 0; k < PT; ++k) {
        int i = k * FPS_THREADS + t;     // coalesced
        float4 p = pts4[i];
        float d = p.w + p0.w - 2.0f * (p.x*p0.x + p.y*p0.y + p.z*p0.z);
        sd[i] = d;
        if (d > mv) { mv = d; mi = i; }
    }
    if (t == 0) fpsIdx[0] = 0;

    for (int s = 1; s < G; ++s) {
        // wave-level argmax (first-occurrence tie-break: lower index wins on equal)
        for (int off = 16; off; off >>= 1) {
            float ov = __shfl_xor(mv, off);
            int   oi = __shfl_xor(mi, off);
            if (ov > mv || (ov == mv && oi < mi)) { mv = ov; mi = oi; }
        }
        if (lane == 0) { wv[wid] = mv; wi[wid] = mi; }
        __syncthreads();
        if (wid == 0) {
            float v2 = wv[lane]; int i2 = wi[lane];
            for (int off = 16; off; off >>= 1) {
                float ov = __shfl_xor(v2, off);
                int   oi = __shfl_xor(i2, off);
                if (ov > v2 || (ov == v2 && oi < i2)) { v2 = ov; i2 = oi; }
            }
            if (lane == 0) { *s_nxt = i2; fpsIdx[s] = i2; }
        }
        __syncthreads();
        int nxt = *s_nxt;
        float4 q = pts4[nxt];
        mv = -1.0f; mi = 0;
        for (int k = 0; k < PT; ++k) {
            int i = k * FPS_THREADS + t;
            float4 p = pts4[i];
            float d  = p.w + q.w - 2.0f * (p.x*q.x + p.y*q.y + p.z*q.z);
            float nd = fminf(sd[i], d);
            sd[i] = nd;
            if (nd > mv) { mv = nd; mi = i; }
        }
    }
}

// ---------------- centers: gather + emit center / fps_idx outputs ----------------
__global__ void center_kernel(const float4* __restrict__ pts4, const int* __restrict__ fpsIdx,
                              float4* __restrict__ cent4,
                              float* __restrict__ outCenter, float* __restrict__ outFps, int G) {
    int g = blockIdx.x * blockDim.x + threadIdx.x;
    if (g < G) {
        int id = fpsIdx[g];
        float4 p = pts4[id];
        cent4[g] = p;
        outCenter[3*g+0] = p.x;
        outCenter[3*g+1] = p.y;
        outCenter[3*g+2] = p.z;
        outFps[g] = (float)id;
    }
}

// ---------------- KNN: WMMA score tiles + wave32 top-32 selection ----------------
__global__ __launch_bounds__(KNN_THREADS) void knn_kernel(const float4* __restrict__ pts4,
                                                          const float4* __restrict__ cent4,
                                                          float* __restrict__ outNbr,
                                                          float* __restrict__ outIdx,
                                                          int N, int G) {
    __shared__ float  sc[CHUNK * SROW];
    __shared__ float4 scent[16];

    const int tid   = threadIdx.x;
    const int lane  = tid & 31;
    const int wave  = tid >> 5;          // 16 waves; wave w owns center gbase+w
    const int gbase = blockIdx.x * 16;

    if (tid < 16) scent[tid] = cent4[gbase + tid];
    __syncthreads();

    // B fragment (4x16, K=4): column g = [-2cx,-2cy,-2cz,1]; loop-invariant.
    // Layout: lanes0-15 -> N=lane, VGPR0=K0, VGPR1=K1; lanes16-31 -> N=lane-16, VGPR0=K2, VGPR1=K3
    float4 cB = scent[lane & 15];
    v2f bfrag;
    bfrag.x = (lane < 16) ? (-2.0f * cB.x) : (-2.0f * cB.z);
    bfrag.y = (lane < 16) ? (-2.0f * cB.y) : 1.0f;

    // sorted-across-lanes top-32 list: lane l holds l-th smallest (value, index)
    float lv = 3.0e38f;
    int   li = 0;

    const int nChunks = N / CHUNK;
    for (int ci = 0; ci < nChunks; ++ci) {
        // prefetch next chunk's points into cache while this chunk computes/selects
        if (ci + 1 < nChunks)
            __builtin_prefetch(&pts4[(ci + 1) * CHUNK + tid], 0, 0);

        // --- WMMA phase: each wave computes 2 of the 32 row-tiles ---
        #pragma unroll
        for (int tt = 0; tt < 2; ++tt) {
            int trow = wave * 2 + tt;                       // local tile row 0..31
            float4 p = pts4[ci * CHUNK + trow * 16 + (lane & 15)];
            // A fragment (16x4): row M = [x,y,z,||x||^2]
            v2f afrag;
            afrag.x = (lane < 16) ? p.x : p.z;              // K0 / K2
            afrag.y = (lane < 16) ? p.y : p.w;              // K1 / K3
            v8f acc = {0.f,0.f,0.f,0.f,0.f,0.f,0.f,0.f};
            acc = __builtin_amdgcn_wmma_f32_16x16x4_f32(
                false, afrag, false, bfrag, (short)0, acc, false, false);
            // D layout: VGPR r -> M=r (lanes0-15) / M=r+8 (lanes16-31), N = lane&15
            int row0 = trow * 16 + ((lane >> 4) << 3);
            int col  = lane & 15;
            #pragma unroll
            for (int r = 0; r < 8; ++r)
                sc[(row0 + r) * SROW + col] = acc[r];
        }
        __syncthreads();

        // --- selection phase: wave w scans column w, 32 candidates at a time ---
        for (int b = 0; b < CHUNK / 32; ++b) {
            int   m     = b * 32 + lane;
            float cand  = sc[m * SROW + wave];
            int   pbase = ci * CHUNK + b * 32;
            unsigned done = 0u;
            while (true) {
                float thr = readlane_f32(lv, 31);                     // v_readlane (uniform)
                unsigned msk = (unsigned)__ballot(cand < thr) & ~done;
                if (msk == 0u) break;
                int j = __builtin_ctz(msk);                           // wave-uniform
                done |= (1u << j);
                float cv   = readlane_f32(cand, j);                   // v_readlane (uniform j)
                int   cidx = pbase + j;
                // stable insert position (existing equal values came from smaller indices)
                unsigned le = (unsigned)__ballot(lv <= cv);
                int pos = __builtin_popcount(le);
                float pv = __shfl_up(lv, 1);
                int   pi = __shfl_up(li, 1);
                if      (lane == pos) { lv = cv; li = cidx; }
                else if (lane >  pos) { lv = pv; li = pi;  }
            }
        }
        __syncthreads();
    }

    // --- write neighborhood (xyz - center) and indices (as float) ---
    int g = gbase + wave;
    float4 cc = scent[wave];
    float4 p  = pts4[li];
    int o = g * 32 + lane;
    outNbr[3*o+0] = p.x - cc.x;
    outNbr[3*o+1] = p.y - cc.y;
    outNbr[3*o+2] = p.z - cc.z;
    outIdx[o] = (float)li;
}

extern "C" void kernel_launch(void* const* d_in, const int* in_sizes, int n_in,
                              void* d_out, int out_size, void* d_ws, size_t ws_size,
                              hipStream_t stream) {
    const float* xyz = (const float*)d_in[0];
    const int N = in_sizes[0] / 3;   // 32768
    const int G = 1024;              // num_group (host-side constant per setup)
    const int S = 32;                // group_size

    // workspace layout
    float4* pts4   = (float4*)d_ws;                       // N * 16B
    int*    fpsIdx = (int*)(pts4 + N);                    // G * 4B
    float4* cent4  = (float4*)(fpsIdx + G);               // G * 16B (16B-aligned: N*16+4096)

    // output layout: neighborhood | center | ori_idx | fps_idx (all as float)
    float* out       = (float*)d_out;
    float* outNbr    = out;                 // G*S*3
    float* outCenter = outNbr + G * S * 3;  // G*3
    float* outIdx    = outCenter + G * 3;   // G*S
    float* outFps    = outIdx + G * S;      // G

    prep_kernel<<<(N + 255) / 256, 256, 0, stream>>>(xyz, pts4, N);

    size_t fpsLds = (size_t)N * sizeof(float) + 32 * sizeof(float) + 32 * sizeof(int) + 16;
    fps_kernel<<<1, FPS_THREADS, fpsLds, stream>>>(pts4, fpsIdx, N, G);

    center_kernel<<<(G + 255) / 256, 256, 0, stream>>>(pts4, fpsIdx, cent4, outCenter, outFps, G);

    knn_kernel<<<G / 16, KNN_THREADS, 0, stream>>>(pts4, cent4, outNbr, outIdx, N, G);
}